// TrittentionCube_27273042329832
// MI455X (gfx1250) — compile-verified
//
#include <hip/hip_runtime.h>

// B=1, S=192, DIM=512, HEADS=8, DH=64
#define SEQ   192
#define DIM_  512
#define HEADS 8
#define DH_   64
#define PROJC 2560   // 5*HEADS*DH

typedef float v2f __attribute__((ext_vector_type(2)));
typedef float v8f __attribute__((ext_vector_type(8)));
typedef unsigned int u32x4 __attribute__((ext_vector_type(4)));
typedef int i32x4 __attribute__((ext_vector_type(4)));
typedef int i32x8 __attribute__((ext_vector_type(8)));

// D = A(16x4,f32) * B(4x16,f32) + C(16x16,f32)
__device__ __forceinline__ v8f wmma4(v2f a, v2f b, v8f c) {
  return __builtin_amdgcn_wmma_f32_16x16x4_f32(
      false, a, false, b, (short)0, c, false, false);
}

// ---- Tensor Data Mover: 2D tile (f32) global -> LDS --------------------
// dims/tiles/stride in f32 elements. pad!=0: insert 4 DWORDs after every 64
// DWORDs stored (row stride 64 -> LDS row stride 68 floats).
__device__ __forceinline__ void tdm_load_2d(unsigned int lds_off, const void* gptr,
                                            unsigned int dim0, unsigned int dim1,
                                            unsigned int tile0, unsigned int tile1,
                                            unsigned int stride0, int pad) {
  unsigned long long ga = (unsigned long long)(uintptr_t)gptr;
  u32x4 g0;
  g0.x = 1u;                                   // count=1, user mode
  g0.y = lds_off;                              // LDS byte address
  g0.z = (unsigned int)ga;                     // global_addr[31:0]
  g0.w = (unsigned int)((ga >> 32) & 0x1FFFFFFu) | (2u << 30);  // addr[56:32] | type=2
  i32x8 g1;
  unsigned int d0 = (2u << 16);                // data_size=4B, mask=0
  if (pad) d0 |= (1u << 20) | (5u << 22) | (3u << 25);  // pad_en, int=64dw, amt=4dw
  g1[0] = (int)d0;
  g1[1] = (int)((dim0 & 0xFFFFu) << 16);                        // tensor_dim0[15:0]
  g1[2] = (int)(((dim0 >> 16) & 0xFFFFu) | ((dim1 & 0xFFFFu) << 16));
  g1[3] = (int)(((dim1 >> 16) & 0xFFFFu) | ((tile0 & 0xFFFFu) << 16));
  g1[4] = (int)(tile1 & 0xFFFFu);                                // tile_dim1, tile_dim2=0
  g1[5] = (int)stride0;                                          // dim0_stride[31:0]
  g1[6] = 0;                                                     // stride0[47:32], stride1 lo
  g1[7] = 0;
  i32x4 g2; g2.x = 1; g2.y = 0; g2.z = 0; g2.w = 0;              // tensor_dim2=1
  i32x4 g3; g3.x = 0; g3.y = 0; g3.z = 0; g3.w = 0;
#if __has_include(<hip/amd_detail/amd_gfx1250_TDM.h>)
  i32x8 g4; g4[0]=0; g4[1]=0; g4[2]=0; g4[3]=0; g4[4]=0; g4[5]=0; g4[6]=0; g4[7]=0;
  __builtin_amdgcn_tensor_load_to_lds(g0, g1, g2, g3, g4, 0);
#else
  __builtin_amdgcn_tensor_load_to_lds(g0, g1, g2, g3, 0);
#endif
}

// ---------------- LayerNorm: x[192,512] -> xn ----------------
__global__ void k_ln(const float* __restrict__ x, const float* __restrict__ w,
                     const float* __restrict__ b, float* __restrict__ xn) {
  __shared__ float red[256];
  int r = blockIdx.x, t = threadIdx.x;
  const float* xr = x + r * DIM_;
  float v0 = xr[t], v1 = xr[t + 256];
  red[t] = v0 + v1;
  __syncthreads();
  for (int o = 128; o > 0; o >>= 1) { if (t < o) red[t] += red[t + o]; __syncthreads(); }
  float mu = red[0] * (1.0f / 512.0f);
  __syncthreads();
  float d0 = v0 - mu, d1 = v1 - mu;
  red[t] = d0 * d0 + d1 * d1;
  __syncthreads();
  for (int o = 128; o > 0; o >>= 1) { if (t < o) red[t] += red[t + o]; __syncthreads(); }
  float var = red[0] * (1.0f / 512.0f);
  float rinv = rsqrtf(var + 1e-5f);
  xn[r * DIM_ + t]       = d0 * rinv * w[t]       + b[t];
  xn[r * DIM_ + t + 256] = d1 * rinv * w[t + 256] + b[t + 256];
}

// ---------------- proj = xn @ W_abcde^T + bias : [192,2560] ----------------
__global__ void k_proj(const float* __restrict__ xn, const float* __restrict__ W,
                       const float* __restrict__ bias, float* __restrict__ proj) {
  int wave = threadIdx.x >> 5, lane = threadIdx.x & 31;
  int t = blockIdx.x * 8 + wave;            // 12 * 160 tiles
  if (t >= 12 * 160) return;
  int mt = t / 160, nt = t % 160;
  int half = lane >> 4, i16 = lane & 15;
  int row0 = mt * 16, col0 = nt * 16;
  v8f c = {};
  for (int k = 0; k < DIM_; k += 4) {
    int ka = k + half * 2;
    v2f a, b;
    a.x = xn[(row0 + i16) * DIM_ + ka];
    a.y = xn[(row0 + i16) * DIM_ + ka + 1];
    b.x = W[(col0 + i16) * DIM_ + ka];       // B[k,n] = W[n,k]
    b.y = W[(col0 + i16) * DIM_ + ka + 1];
    c = wmma4(a, b, c);
  }
  float bv = bias[col0 + i16];
#pragma unroll
  for (int j = 0; j < 8; ++j)
    proj[(row0 + half * 8 + j) * PROJC + col0 + i16] = c[j] + bv;
}

// ---------------- bsum[r,j] = sum_h b_[r,h,j] ----------------
__global__ void k_bsum(const float* __restrict__ proj, float* __restrict__ bsum) {
  int idx = blockIdx.x * blockDim.x + threadIdx.x;  // 192*64
  if (idx >= SEQ * DH_) return;
  int r = idx >> 6, j = idx & 63;
  const float* p = proj + r * PROJC + 512 + j;
  float s = 0.f;
#pragma unroll
  for (int h = 0; h < HEADS; ++h) s += p[h * 64];
  bsum[idx] = s;
}

// ---------------- step1[r, i*64+j] = sum_{h,k} c_[r,h,k] W_K[h,i,j,k] ----------------
__global__ void k_step1(const float* __restrict__ proj, const float* __restrict__ WK,
                        float* __restrict__ step1) {
  int wave = threadIdx.x >> 5, lane = threadIdx.x & 31;
  int t = blockIdx.x * 8 + wave;            // 12 * 256 tiles
  if (t >= 12 * 256) return;
  int mt = t / 256, nt = t % 256;
  int half = lane >> 4, i16 = lane & 15;
  int row0 = mt * 16, col0 = nt * 16;
  int n = col0 + i16, wi = n >> 6, wj = n & 63;
  v8f c = {};
  for (int k = 0; k < 512; k += 4) {
    int ka = k + half * 2;
    v2f a, b;
    const float* arow = proj + (row0 + i16) * PROJC + 1024;  // c_ section
    a.x = arow[ka];
    a.y = arow[ka + 1];
    int h0 = ka >> 6, k0 = ka & 63;
    int h1 = (ka + 1) >> 6, k1 = (ka + 1) & 63;
    b.x = WK[((h0 * 64 + wi) * 64 + wj) * 64 + k0];
    b.y = WK[((h1 * 64 + wi) * 64 + wj) * 64 + k1];
    c = wmma4(a, b, c);
  }
#pragma unroll
  for (int j = 0; j < 8; ++j)
    step1[(row0 + half * 8 + j) * 4096 + n] = c[j];
}

// ---------------- step2[(r*64+i), q] = sum_j step1[r,i,j] * bsum[q,j] ----------------
__global__ void k_step2(const float* __restrict__ step1, const float* __restrict__ bsum,
                        float* __restrict__ step2) {
  int wave = threadIdx.x >> 5, lane = threadIdx.x & 31;
  int t = blockIdx.x * 8 + wave;            // 768 * 12 tiles
  if (t >= 768 * 12) return;
  int mt = t / 12, nt = t % 12;
  int half = lane >> 4, i16 = lane & 15;
  int row0 = mt * 16, col0 = nt * 16;
  v8f c = {};
  for (int k = 0; k < 64; k += 4) {
    int ka = k + half * 2;
    v2f a, b;
    a.x = step1[(row0 + i16) * 64 + ka];
    a.y = step1[(row0 + i16) * 64 + ka + 1];
    b.x = bsum[(col0 + i16) * 64 + ka];      // B[j,q] = bsum[q,j]
    b.y = bsum[(col0 + i16) * 64 + ka + 1];
    c = wmma4(a, b, c);
  }
#pragma unroll
  for (int j = 0; j < 8; ++j)
    step2[(row0 + half * 8 + j) * 192 + col0 + i16] = c[j];
}

// ---------------- T[h][p][g*64+f] = sum_dh d_[p,h,dh] W_V[h,dh,g,f] ----------------
__global__ void k_tv(const float* __restrict__ proj, const float* __restrict__ WV,
                     float* __restrict__ T) {
  int wave = threadIdx.x >> 5, lane = threadIdx.x & 31;
  int t = blockIdx.x * 8 + wave;            // 8 * 12 * 256 tiles
  if (t >= 8 * 12 * 256) return;
  int h = t / (12 * 256);
  int rem = t % (12 * 256);
  int mt = rem / 256, nt = rem % 256;
  int half = lane >> 4, i16 = lane & 15;
  int row0 = mt * 16, col0 = nt * 16;
  const float* B = WV + h * 262144;         // [dh][4096] row-major
  v8f c = {};
  for (int k = 0; k < 64; k += 4) {
    int ka = k + half * 2;
    v2f a, b;
    const float* arow = proj + (row0 + i16) * PROJC + 1536 + h * 64;  // d_
    a.x = arow[ka];
    a.y = arow[ka + 1];
    b.x = B[ka * 4096 + col0 + i16];
    b.y = B[(ka + 1) * 4096 + col0 + i16];
    c = wmma4(a, b, c);
  }
  float* Th = T + h * 786432;
#pragma unroll
  for (int j = 0; j < 8; ++j)
    Th[(row0 + half * 8 + j) * 4096 + col0 + i16] = c[j];
}

// ---------------- v[h][p][q][f] = sum_g e_[q,h,g] * T[h][p][g*64+f] ----------------
// One block per (h,p); TDM-stage e_ panel (padded) + T[h][p] panel; 8 waves x 6 tiles.
__global__ void __launch_bounds__(256)
k_v(const float* __restrict__ proj, const float* __restrict__ T, float* __restrict__ v) {
  __shared__ float eLds[192 * 68];   // 52.2KB, padded row stride 68
  __shared__ float TLds[64 * 64];    // 16KB
  int wave = threadIdx.x >> 5, lane = threadIdx.x & 31;
  int hp = blockIdx.x;               // 1536
  int h = hp / 192, p = hp % 192;
  int half = lane >> 4, i16 = lane & 15;
  if (wave == 0) {
    tdm_load_2d((unsigned int)(uintptr_t)eLds, proj + 2048 + h * 64,
                64, 192, 64, 192, PROJC, /*pad=*/1);
    tdm_load_2d((unsigned int)(uintptr_t)TLds, T + h * 786432 + p * 4096,
                4096, 1, 4096, 1, 4096, /*pad=*/0);
    __builtin_amdgcn_s_wait_tensorcnt(0);
  }
  __syncthreads();
  float* vout = v + (size_t)hp * (192 * 64);
  for (int tt = wave * 6; tt < wave * 6 + 6; ++tt) {   // 48 tiles: qt(12) x ft(4)
    int qt = tt >> 2, ftl = tt & 3;
    int row0 = qt * 16, col0 = ftl * 16;
    v8f c = {};
    for (int k = 0; k < 64; k += 4) {
      int ka = k + half * 2;
      v2f a, b;
      a.x = eLds[(row0 + i16) * 68 + ka];
      a.y = eLds[(row0 + i16) * 68 + ka + 1];
      b.x = TLds[ka * 64 + col0 + i16];
      b.y = TLds[(ka + 1) * 64 + col0 + i16];
      c = wmma4(a, b, c);
    }
#pragma unroll
    for (int j = 0; j < 8; ++j)
      vout[(row0 + half * 8 + j) * 64 + col0 + i16] = c[j];
  }
}

// ---------------- Fused flash attention partials ----------------
// grid = h(8) x rt(12) x qt(12) = 1152 blocks, 8 waves; wave owns 2 r rows.
// Online softmax over this block's 192p x 16q slice; partial z + (m,s) out.
__global__ void __launch_bounds__(256)
k_attn1(const float* __restrict__ proj, const float* __restrict__ step2,
        const float* __restrict__ v, float* __restrict__ zpart,
        float* __restrict__ mpart, float* __restrict__ spart) {
  __shared__ float aLds[192 * 68];       // 52.2KB a_ panel (padded)
  __shared__ float Blds[16 * 64 * 16];   // 64KB: 16 r-panels [64k x 16q]
  __shared__ float Wlds[16 * 257];       // 16.4KB weight chunk [16r x 256l] (padded)
  __shared__ float scaleLds[16];
  __shared__ float zred[4 * 256];        // khalf reduction

  int wave = threadIdx.x >> 5, lane = threadIdx.x & 31;
  int b = blockIdx.x;
  int h = b / 144, rt = (b / 12) % 12, qt = b % 12;
  int r0 = rt * 16;
  int half = lane >> 4, i16 = lane & 15;

  // --- TDM staging: a_ head panel (wave0) + per-wave step2 r-panels ---
  if (wave == 0) {
    tdm_load_2d((unsigned int)(uintptr_t)aLds, proj + h * 64,
                64, 192, 64, 192, PROJC, /*pad=*/1);
  }
  for (int rr = 0; rr < 2; ++rr) {
    int rloc = wave * 2 + rr;
    tdm_load_2d((unsigned int)(uintptr_t)(&Blds[rloc * 1024]),
                step2 + (r0 + rloc) * 64 * 192 + qt * 16,
                16, 64, 16, 64, 192, /*pad=*/0);
  }
  __builtin_amdgcn_s_wait_tensorcnt(0);
  __syncthreads();

  const float* vh = v + (size_t)h * (192 * 192 * 64);
  int ft = wave & 3, khalf = wave >> 2;
  float m_run[2] = {-3.0e38f, -3.0e38f};
  float s_run[2] = {0.0f, 0.0f};
  v8f zacc = {};

  for (int pt = 0; pt < 12; ++pt) {
    // 1) logits for my 2 r rows, online (m,s), weights -> Wlds
    for (int rr = 0; rr < 2; ++rr) {
      int rloc = wave * 2 + rr;
      const float* Bl = &Blds[rloc * 1024];
      v8f c = {};
      for (int ks = 0; ks < 16; ++ks) {
        int ka = ks * 4 + half * 2;
        v2f a, bf;
        a.x = aLds[(pt * 16 + i16) * 68 + ka];
        a.y = aLds[(pt * 16 + i16) * 68 + ka + 1];
        bf.x = Bl[ka * 16 + i16];
        bf.y = Bl[(ka + 1) * 16 + i16];
        c = wmma4(a, bf, c);
      }
      float lg[8];
      float tm = -3.0e38f;
#pragma unroll
      for (int j = 0; j < 8; ++j) {
        int p = pt * 16 + half * 8 + j, q = qt * 16 + i16;
        float val = (p >= q) ? 1e-6f : c[j];
        lg[j] = val * (1.0f / 64.0f);
        tm = fmaxf(tm, lg[j]);
      }
      for (int off = 16; off > 0; off >>= 1)
        tm = fmaxf(tm, __shfl_xor(tm, off, 32));
      float nm = fmaxf(m_run[rr], tm);
      float ls = 0.0f;
#pragma unroll
      for (int j = 0; j < 8; ++j) {
        float w = __expf(lg[j] - nm);
        ls += w;
        Wlds[rloc * 257 + (half * 8 + j) * 16 + i16] = w;
      }
      for (int off = 16; off > 0; off >>= 1)
        ls += __shfl_xor(ls, off, 32);
      float scale = __expf(m_run[rr] - nm);
      s_run[rr] = s_run[rr] * scale + ls;
      m_run[rr] = nm;
      if (lane == 0) scaleLds[rloc] = scale;
    }
    __syncthreads();
    // 2) rescale accumulator rows, accumulate chunk: zacc += W(16x256) @ v(256x16)
#pragma unroll
    for (int j = 0; j < 8; ++j) zacc[j] *= scaleLds[half * 8 + j];
    for (int ks = khalf * 32; ks < khalf * 32 + 32; ++ks) {
      int k0 = ks * 4 + half * 2;
      v2f a2, b2;
      a2.x = Wlds[i16 * 257 + k0];
      a2.y = Wlds[i16 * 257 + k0 + 1];
      int l0 = (pt * 16 + (k0 >> 4)) * 192 + qt * 16 + (k0 & 15);
      int l1 = (pt * 16 + ((k0 + 1) >> 4)) * 192 + qt * 16 + ((k0 + 1) & 15);
      b2.x = vh[l0 * 64 + ft * 16 + i16];
      b2.y = vh[l1 * 64 + ft * 16 + i16];
      zacc = wmma4(a2, b2, zacc);
    }
    __syncthreads();
  }

  // combine khalf halves; store partial z, (m,s)
  if (khalf == 1) {
#pragma unroll
    for (int j = 0; j < 8; ++j)
      zred[ft * 256 + (half * 8 + j) * 16 + i16] = zacc[j];
  }
  __syncthreads();
  if (khalf == 0) {
#pragma unroll
    for (int j = 0; j < 8; ++j) {
      float val = zacc[j] + zred[ft * 256 + (half * 8 + j) * 16 + i16];
      zpart[(size_t)b * 1024 + (half * 8 + j) * 64 + ft * 16 + i16] = val;
    }
  }
  if (lane == 0) {
    for (int rr = 0; rr < 2; ++rr) {
      int rloc = wave * 2 + rr;
      mpart[(h * 192 + r0 + rloc) * 12 + qt] = m_run[rr];
      spart[(h * 192 + r0 + rloc) * 12 + qt] = s_run[rr];
    }
  }
}

// ---------------- combine q-partials: z[r, h*64+f] ----------------
__global__ void k_attn2(const float* __restrict__ zpart, const float* __restrict__ mpart,
                        const float* __restrict__ spart, float* __restrict__ z) {
  int wave = threadIdx.x >> 5, lane = threadIdx.x & 31;
  int idx = blockIdx.x * 8 + wave;     // h*192 + r, 1536 total
  if (idx >= 8 * 192) return;
  int h = idx / 192, r = idx % 192;
  int rt = r / 16, rloc = r % 16;
  const float* mrow = mpart + idx * 12;
  const float* srow = spart + idx * 12;
  float M = -3.0e38f;
  for (int qt = 0; qt < 12; ++qt) M = fmaxf(M, mrow[qt]);
  float S = 0.0f;
  for (int qt = 0; qt < 12; ++qt) S += srow[qt] * __expf(mrow[qt] - M);
  float inv = 1.0f / S;
  float acc0 = 0.0f, acc1 = 0.0f;
  for (int qt = 0; qt < 12; ++qt) {
    float sc = __expf(mrow[qt] - M);
    const float* zp = zpart + (size_t)((h * 12 + rt) * 12 + qt) * 1024 + rloc * 64;
    acc0 += zp[lane] * sc;
    acc1 += zp[lane + 32] * sc;
  }
  z[r * 512 + h * 64 + lane]      = acc0 * inv;
  z[r * 512 + h * 64 + lane + 32] = acc1 * inv;
}

// ---------------- out = z @ W_out^T + b_out : [192,512] ----------------
__global__ void k_out(const float* __restrict__ z, const float* __restrict__ Wo,
                      const float* __restrict__ bo, float* __restrict__ out) {
  int wave = threadIdx.x >> 5, lane = threadIdx.x & 31;
  int t = blockIdx.x * 8 + wave;            // 12 * 32 tiles
  if (t >= 12 * 32) return;
  int mt = t / 32, nt = t % 32;
  int half = lane >> 4, i16 = lane & 15;
  int row0 = mt * 16, col0 = nt * 16;
  v8f c = {};
  for (int k = 0; k < 512; k += 4) {
    int ka = k + half * 2;
    v2f a, b;
    a.x = z[(row0 + i16) * 512 + ka];
    a.y = z[(row0 + i16) * 512 + ka + 1];
    b.x = Wo[(col0 + i16) * 512 + ka];
    b.y = Wo[(col0 + i16) * 512 + ka + 1];
    c = wmma4(a, b, c);
  }
  float bv = bo[col0 + i16];
#pragma unroll
  for (int j = 0; j < 8; ++j)
    out[(row0 + half * 8 + j) * 512 + col0 + i16] = c[j] + bv;
}

extern "C" void kernel_launch(void* const* d_in, const int* in_sizes, int n_in,
                              void* d_out, int out_size, void* d_ws, size_t ws_size,
                              hipStream_t stream) {
  (void)in_sizes; (void)n_in; (void)out_size; (void)ws_size;
  const float* x       = (const float*)d_in[0];
  const float* ln_w    = (const float*)d_in[1];
  const float* ln_b    = (const float*)d_in[2];
  const float* W_abcde = (const float*)d_in[3];
  const float* b_abcde = (const float*)d_in[4];
  const float* W_K     = (const float*)d_in[5];
  const float* W_V     = (const float*)d_in[6];
  const float* W_out   = (const float*)d_in[7];
  const float* b_out   = (const float*)d_in[8];
  float* out = (float*)d_out;

  float* ws    = (float*)d_ws;
  float* xn    = ws;                 //    98304
  float* proj  = ws + 98304;         //   491520
  float* bsum  = ws + 589824;        //    12288
  float* step1 = ws + 602112;        //   786432
  float* step2 = ws + 1388544;       //  2359296
  float* T     = ws + 3747840;       //  6291456
  float* v     = ws + 10039296;      // 18874368
  float* z     = ws + 28913664;      //    98304
  float* zpart = ws + 29011968;      //  1179648
  float* mpart = ws + 30191616;      //    18432
  float* spart = ws + 30210048;      //    18432

  k_ln   <<<192,  256, 0, stream>>>(x, ln_w, ln_b, xn);
  k_proj <<<240,  256, 0, stream>>>(xn, W_abcde, b_abcde, proj);
  k_bsum <<<48,   256, 0, stream>>>(proj, bsum);
  k_step1<<<384,  256, 0, stream>>>(proj, W_K, step1);
  k_step2<<<1152, 256, 0, stream>>>(step1, bsum, step2);
  k_tv   <<<3072, 256, 0, stream>>>(proj, W_V, T);
  k_v    <<<1536, 256, 0, stream>>>(proj, T, v);
  k_attn1<<<1152, 256, 0, stream>>>(proj, step2, v, zpart, mpart, spart);
  k_attn2<<<192,  256, 0, stream>>>(zpart, mpart, spart, z);
  k_out  <<<48,   256, 0, stream>>>(z, W_out, b_out, out);
}